// MemoryDecoder_86870008529728
// MI455X (gfx1250) — compile-verified
//
#include <hip/hip_runtime.h>
#include <hip/hip_bf16.h>

#define HW   3072
#define H1   48
#define W1   64
#define NK   (HW * 48)   // cost_memory rows
#define DEPTH 6

typedef float v2f __attribute__((ext_vector_type(2)));
typedef float v8f __attribute__((ext_vector_type(8)));

// ---------------------------------------------------------------------------
// fp32 WMMA wrapper: V_WMMA_F32_16X16X4_F32  (D = A(16x4) * B(4x16) + C)
// ---------------------------------------------------------------------------
__device__ __forceinline__ v8f wmma_f32_16x16x4(v2f a, v2f b, v8f c) {
#if __has_builtin(__builtin_amdgcn_wmma_f32_16x16x4_f32)
  return __builtin_amdgcn_wmma_f32_16x16x4_f32(false, a, false, b, (short)0, c,
                                               false, false);
#else
  c[0] += a.x * b.x + a.y * b.y;   // keep TU compiling if builtin absent
  return c;
#endif
}

// ---------------------------------------------------------------------------
// Generic strided GEMM:  C(m,n) = act(alpha * (sum_k A(m,k)B(k,n) + bias[n]))
// A(m,k)=A[m*sAm+k*sAk], B(k,n)=B[k*sBk+n*sBn], C(m,n)=C[m*sCm+n*sCn]
// One wave -> one 16x16 C tile via V_WMMA_F32_16X16X4_F32, K stepped by 4.
// ASSUMES M%16==0 and N%16==0 (true for every call in this model); only K
// may be ragged (81, 145) and is handled by one guarded tail iteration.
// Main loop is branch-free + software-pipelined (loads for k+8 issued before
// the two WMMAs for k), with L2 prefetch ~56 K-steps ahead.
// act: 0 none, 1 relu, 2 gelu(exact), 3 tanh, 4 sigmoid
// ---------------------------------------------------------------------------
__global__ void __launch_bounds__(128) gemm_wmma_f32(
    const float* __restrict__ A, long sAm, long sAk,
    const float* __restrict__ B, long sBk, long sBn,
    float* __restrict__ C, long sCm, long sCn,
    const float* __restrict__ bias,
    int M, int N, int K, float alpha, int act)
{
  const int tilesN = N >> 4;
  const int tilesM = M >> 4;
  const int tile   = blockIdx.x * 4 + (threadIdx.x >> 5);
  if (tile >= tilesM * tilesN) return;          // wave-uniform exit
  const int m0 = (tile / tilesN) << 4;
  const int n0 = (tile % tilesN) << 4;
  const int lane = threadIdx.x & 31;
  const int l2 = lane & 15;
  const int hi = lane >> 4;

  // lane's base pointers: A row m0+l2 at k=2*hi ; B col n0+l2 at k=2*hi
  const float* Ap0 = A + (long)(m0 + l2) * sAm + (long)(2 * hi) * sAk;
  const float* Bp0 = B + (long)(2 * hi) * sBk + (long)(n0 + l2) * sBn;
#define LDA(k) Ap0[(long)(k) * sAk]
#define LDB(k) Bp0[(long)(k) * sBk]

  v8f acc = {0.f,0.f,0.f,0.f,0.f,0.f,0.f,0.f};
  const int Kmain = K & ~7;
  int k0 = 0;
  if (Kmain) {
    v2f a0 = {LDA(0), LDA(1)}, b0 = {LDB(0), LDB(1)};
    v2f a1 = {LDA(4), LDA(5)}, b1 = {LDB(4), LDB(5)};
    for (k0 = 8; k0 < Kmain; k0 += 8) {
      v2f a2 = {LDA(k0),     LDA(k0 + 1)}, b2 = {LDB(k0),     LDB(k0 + 1)};
      v2f a3 = {LDA(k0 + 4), LDA(k0 + 5)}, b3 = {LDB(k0 + 4), LDB(k0 + 5)};
      __builtin_prefetch(&LDA(k0 + 56), 0, 1);
      __builtin_prefetch(&LDB(k0 + 56), 0, 1);
      acc = wmma_f32_16x16x4(a0, b0, acc);
      acc = wmma_f32_16x16x4(a1, b1, acc);
      a0 = a2; b0 = b2; a1 = a3; b1 = b3;
    }
    acc = wmma_f32_16x16x4(a0, b0, acc);
    acc = wmma_f32_16x16x4(a1, b1, acc);
    k0 = Kmain;
  }
  if (k0 + 4 <= K) {                  // unguarded 4-step (K%8 in [4,7])
    v2f a = {LDA(k0), LDA(k0 + 1)}, b = {LDB(k0), LDB(k0 + 1)};
    acc = wmma_f32_16x16x4(a, b, acc);
    k0 += 4;
  }
  if (k0 < K) {                       // ragged tail (K%4 != 0), guarded once
    const int ka = k0 + 2 * hi;
    v2f a = {0.f, 0.f}, b = {0.f, 0.f};
    if (ka < K)     { a.x = LDA(k0);     b.x = LDB(k0); }
    if (ka + 1 < K) { a.y = LDA(k0 + 1); b.y = LDB(k0 + 1); }
    acc = wmma_f32_16x16x4(a, b, acc);
  }
#undef LDA
#undef LDB

  // epilogue: lane l2 holds column n0+l2, rows m0 + r + 8*hi
#pragma unroll
  for (int r = 0; r < 8; ++r) {
    const int m = m0 + r + 8 * hi;
    const int n = n0 + l2;
    float v = acc[r];
    if (bias) v += bias[n];
    v *= alpha;
    switch (act) {
      case 1: v = fmaxf(v, 0.f); break;
      case 2: v = 0.5f * v * (1.f + erff(v * 0.70710678118654752f)); break;
      case 3: v = tanhf(v); break;
      case 4: v = 1.f / (1.f + expf(-v)); break;
      default: break;
    }
    C[(long)m * sCm + (long)n * sCn] = v;
  }
}

// ---------------------------------------------------------------------------
// Direct conv2d, channel-major (C,H,W) in/out, optional two-part concat input
// act: 0 none, 1 relu, 3 tanh, 4 sigmoid
// ---------------------------------------------------------------------------
__global__ void conv2d_k(const float* __restrict__ inA, int CA,
                         const float* __restrict__ inB, int CB,
                         const float* __restrict__ w,
                         const float* __restrict__ bias,
                         float* __restrict__ out, int Cout,
                         int KH, int KW, int PH, int PW, int act)
{
  const int idx = blockIdx.x * blockDim.x + threadIdx.x;
  if (idx >= Cout * HW) return;
  const int o = idx / HW, pix = idx % HW;
  const int y = pix >> 6, x = pix & 63;
  const int Cin = CA + CB;
  float acc = bias ? bias[o] : 0.f;
  const float* wo = w + (long)o * Cin * KH * KW;
  for (int c = 0; c < Cin; ++c) {
    const float* src = (c < CA) ? (inA + (long)c * HW) : (inB + (long)(c - CA) * HW);
    const float* wc  = wo + (long)c * KH * KW;
    for (int ky = 0; ky < KH; ++ky) {
      const int iy = y + ky - PH;
      if ((unsigned)iy >= (unsigned)H1) continue;
      for (int kx = 0; kx < KW; ++kx) {
        const int ix = x + kx - PW;
        if ((unsigned)ix >= (unsigned)W1) continue;
        acc += src[iy * W1 + ix] * wc[ky * KW + kx];
      }
    }
  }
  switch (act) {
    case 1: acc = fmaxf(acc, 0.f); break;
    case 3: acc = tanhf(acc); break;
    case 4: acc = 1.f / (1.f + expf(-acc)); break;
    default: break;
  }
  out[(long)o * HW + pix] = acc;
}

// ---------------------------------------------------------------------------
// Row softmax (in place), one block per row
// ---------------------------------------------------------------------------
__global__ void __launch_bounds__(256) softmax_rows(float* __restrict__ a, int ncols)
{
  __shared__ float red[256];
  const int row = blockIdx.x, t = threadIdx.x;
  float* r = a + (long)row * ncols;
  float mx = -1e30f;
  for (int c = t; c < ncols; c += 256) mx = fmaxf(mx, r[c]);
  red[t] = mx; __syncthreads();
  for (int o = 128; o > 0; o >>= 1) { if (t < o) red[t] = fmaxf(red[t], red[t + o]); __syncthreads(); }
  mx = red[0]; __syncthreads();
  float sm = 0.f;
  for (int c = t; c < ncols; c += 256) { float e = expf(r[c] - mx); r[c] = e; sm += e; }
  red[t] = sm; __syncthreads();
  for (int o = 128; o > 0; o >>= 1) { if (t < o) red[t] += red[t + o]; __syncthreads(); }
  const float inv = 1.f / red[0];
  for (int c = t; c < ncols; c += 256) r[c] *= inv;
}

// ---------------------------------------------------------------------------
// Small block-of-64 reduce helper
// ---------------------------------------------------------------------------
__device__ __forceinline__ float block_sum64(float v, float* red, int t) {
  red[t] = v; __syncthreads();
  for (int o = 32; o > 0; o >>= 1) { if (t < o) red[t] += red[t + o]; __syncthreads(); }
  float s = red[0]; __syncthreads();
  return s;
}

// layernorm(query)*g+b + positional embedding of coords1
__global__ void __launch_bounds__(64) k_ln_pe(
    const float* __restrict__ q, const float* __restrict__ g,
    const float* __restrict__ b, const float* __restrict__ c1,
    float* __restrict__ out)
{
  __shared__ float red[64];
  const int p = blockIdx.x, t = threadIdx.x;
  const float v = q[p * 64 + t];
  const float mean = block_sum64(v, red, t) * (1.f / 64.f);
  const float d = v - mean;
  const float var = block_sum64(d * d, red, t) * (1.f / 64.f);
  const float qn = d * rsqrtf(var + 1e-5f) * g[t] + b[t];
  const float x = c1[p], y = c1[HW + p];
  const int grp = t >> 4, i = t & 15;
  const float a = 3.14f * (float)i * (1.0f / 200.0f);
  const float ang = (grp < 2 ? x : y) * a;
  const float pe = (grp & 1) ? cosf(ang) : sinf(ang);
  out[p * 64 + t] = qn + pe;
}

// x = query + projout ; xn = LN(x)*g+b
__global__ void __launch_bounds__(64) k_x_xn(
    const float* __restrict__ query, const float* __restrict__ projout,
    const float* __restrict__ g, const float* __restrict__ b,
    float* __restrict__ xb, float* __restrict__ xn)
{
  __shared__ float red[64];
  const int p = blockIdx.x, t = threadIdx.x;
  const float v = query[p * 64 + t] + projout[p * 64 + t];
  xb[p * 64 + t] = v;
  const float mean = block_sum64(v, red, t) * (1.f / 64.f);
  const float d = v - mean;
  const float var = block_sum64(d * d, red, t) * (1.f / 64.f);
  xn[p * 64 + t] = d * rsqrtf(var + 1e-5f) * g[t] + b[t];
}

// ---------------------------------------------------------------------------
// Cross-attention core: per pixel, 8 heads x 48 tokens x d=8
// Writes cat128[p] = [attn_out(64) | query(64)]
// ---------------------------------------------------------------------------
__global__ void __launch_bounds__(384) attn_core(
    const float* __restrict__ qp, const float* __restrict__ key,
    const float* __restrict__ val, const float* __restrict__ query,
    float* __restrict__ cat128)
{
  __shared__ float s[8][48];
  __shared__ float hs[8];
  const int p = blockIdx.x, t = threadIdx.x;
  const int h = t / 48, j = t % 48;
  {
    const float* kp = key + ((long)p * 48 + j) * 64 + h * 8;
    const float* qq = qp + (long)p * 64 + h * 8;
    float dot = 0.f;
#pragma unroll
    for (int d = 0; d < 8; ++d) dot += qq[d] * kp[d];
    s[h][j] = dot * 0.35355339059327373f;  // (QD/HEADS)^-0.5
  }
  __syncthreads();
  if (t < 8) {
    float mx = -1e30f;
    for (int jj = 0; jj < 48; ++jj) mx = fmaxf(mx, s[t][jj]);
    float sm = 0.f;
    for (int jj = 0; jj < 48; ++jj) { float e = expf(s[t][jj] - mx); s[t][jj] = e; sm += e; }
    hs[t] = 1.f / sm;
  }
  __syncthreads();
  if (t < 64) {
    const int hh = t >> 3, dd = t & 7;
    const float inv = hs[hh];
    const float* vp = val + (long)p * 48 * 64 + hh * 8 + dd;
    float acc = 0.f;
    for (int jj = 0; jj < 48; ++jj) acc += s[hh][jj] * vp[jj * 64];
    cat128[(long)p * 128 + t] = acc * inv;
    cat128[(long)p * 128 + 64 + t] = query[p * 64 + t];
  }
}

// ---------------------------------------------------------------------------
// Cost-volume bilinear sampling (81 taps per pixel) -> corr channels [64..145)
// ---------------------------------------------------------------------------
__device__ __forceinline__ float bil_tap(const float* img, float x, float y, float w) {
  if (x >= 0.f && x <= 63.f && y >= 0.f && y <= 47.f)
    return img[(int)y * W1 + (int)x] * w;
  return 0.f;
}

__global__ void cf_sample(const float* __restrict__ cm,
                          const float* __restrict__ c1,
                          float* __restrict__ corr)
{
  const int idx = blockIdx.x * blockDim.x + threadIdx.x;
  if (idx >= 81 * HW) return;
  const int c = idx / HW, p = idx % HW;
  const int i = c / 9, j = c % 9;     // channel = i*9+j ; x += d[i], y += d[j]
  const float xs = c1[p] + (float)(i - 4);
  const float ys = c1[HW + p] + (float)(j - 4);
  const float x0 = floorf(xs), y0 = floorf(ys);
  const float wx = xs - x0, wy = ys - y0;
  const float* img = cm + (long)p * HW;
  float acc = 0.f;
  acc += bil_tap(img, x0,       y0,       (1.f - wx) * (1.f - wy));
  acc += bil_tap(img, x0 + 1.f, y0,       wx * (1.f - wy));
  acc += bil_tap(img, x0,       y0 + 1.f, (1.f - wx) * wy);
  acc += bil_tap(img, x0 + 1.f, y0 + 1.f, wx * wy);
  corr[(long)(64 + c) * HW + p] = acc;
}

// ---------------------------------------------------------------------------
// Convex upsampling: one thread per (pixel, 8x8 sub-position)
// ---------------------------------------------------------------------------
__global__ void upsample_k(const float* __restrict__ mask,
                           const float* __restrict__ c1,
                           float* __restrict__ out)
{
  const int idx = blockIdx.x * blockDim.x + threadIdx.x;
  if (idx >= HW * 64) return;
  const int pix = idx >> 6, ab = idx & 63;
  const int y = pix >> 6, x = pix & 63;
  float lg[9];
  float mx = -1e30f;
#pragma unroll
  for (int k = 0; k < 9; ++k) {
    lg[k] = mask[(long)(k * 64 + ab) * HW + pix];
    mx = fmaxf(mx, lg[k]);
  }
  float sm = 0.f;
#pragma unroll
  for (int k = 0; k < 9; ++k) { lg[k] = expf(lg[k] - mx); sm += lg[k]; }
  const float inv = 1.f / sm;
  float a0 = 0.f, a1 = 0.f;
#pragma unroll
  for (int k = 0; k < 9; ++k) {
    const int ki = k / 3, kj = k % 3;
    const int yy = y + ki - 1, xx = x + kj - 1;
    if ((unsigned)yy < (unsigned)H1 && (unsigned)xx < (unsigned)W1) {
      const int pp = yy * W1 + xx;
      const float m = lg[k] * inv;
      a0 += m * 8.f * (c1[pp] - (float)xx);
      a1 += m * 8.f * (c1[HW + pp] - (float)yy);
    }
  }
  const int a = ab >> 3, b = ab & 7;
  const long Y = (long)(y * 8 + a), X = (long)(x * 8 + b);
  out[Y * 512 + X] = a0;
  out[(long)384 * 512 + Y * 512 + X] = a1;
}

// ---------------------------------------------------------------------------
// Misc elementwise kernels
// ---------------------------------------------------------------------------
__global__ void k_init_coords(float* c1) {
  const int t = blockIdx.x * blockDim.x + threadIdx.x;
  if (t >= HW) return;
  c1[t] = (float)(t & 63);
  c1[HW + t] = (float)(t >> 6);
}
__global__ void k_flow(const float* __restrict__ c1, float* __restrict__ flowbuf,
                       float* __restrict__ xcat) {
  const int t = blockIdx.x * blockDim.x + threadIdx.x;
  if (t >= HW) return;
  const float fx = c1[t] - (float)(t & 63);
  const float fy = c1[HW + t] - (float)(t >> 6);
  flowbuf[t] = fx;  flowbuf[HW + t] = fy;
  xcat[(long)254 * HW + t] = fx;          // mf channel 126
  xcat[(long)255 * HW + t] = fy;          // mf channel 127
}
__global__ void k_cost_global(const float* __restrict__ xb,
                              const float* __restrict__ ffno,
                              float* __restrict__ corr) {
  const int idx = blockIdx.x * blockDim.x + threadIdx.x;
  if (idx >= 64 * HW) return;
  const int c = idx / HW, p = idx % HW;
  corr[idx] = xb[p * 64 + c] + ffno[p * 64 + c];
}
__global__ void k_mfg(float* __restrict__ xcat, const float* __restrict__ glob,
                      const float* __restrict__ gamma) {
  const int idx = blockIdx.x * blockDim.x + threadIdx.x;
  if (idx >= 128 * HW) return;
  xcat[(long)256 * HW + idx] = xcat[(long)128 * HW + idx] + gamma[0] * glob[idx];
}
__global__ void k_mul(const float* __restrict__ a, const float* __restrict__ b,
                      float* __restrict__ o, int n) {
  const int i = blockIdx.x * blockDim.x + threadIdx.x;
  if (i < n) o[i] = a[i] * b[i];
}
__global__ void k_gru(float* __restrict__ h, const float* __restrict__ z,
                      const float* __restrict__ q, int n) {
  const int i = blockIdx.x * blockDim.x + threadIdx.x;
  if (i < n) h[i] = (1.f - z[i]) * h[i] + z[i] * q[i];
}
__global__ void k_addcoords(float* __restrict__ c1, const float* __restrict__ df) {
  const int i = blockIdx.x * blockDim.x + threadIdx.x;
  if (i < 2 * HW) c1[i] += df[i];
}

// ---------------------------------------------------------------------------
// Host side
// ---------------------------------------------------------------------------
static void gemm(hipStream_t s, const float* A, long sAm, long sAk,
                 const float* B, long sBk, long sBn,
                 float* C, long sCm, long sCn, const float* bias,
                 int M, int N, int K, float alpha, int act)
{
  const int tiles = (M / 16) * (N / 16);
  gemm_wmma_f32<<<(tiles + 3) / 4, 128, 0, s>>>(A, sAm, sAk, B, sBk, sBn,
                                                C, sCm, sCn, bias, M, N, K,
                                                alpha, act);
}
static void conv(hipStream_t s, const float* inA, int CA, const float* inB, int CB,
                 const float* w, const float* bias, float* out, int Cout,
                 int KH, int KW, int PH, int PW, int act)
{
  const int n = Cout * HW;
  conv2d_k<<<(n + 255) / 256, 256, 0, s>>>(inA, CA, inB, CB, w, bias, out,
                                           Cout, KH, KW, PH, PW, act);
}
#define L1D(kern, n, ...) kern<<<((n) + 255) / 256, 256, 0, stream>>>(__VA_ARGS__)

extern "C" void kernel_launch(void* const* d_in, const int* in_sizes, int n_in,
                              void* d_out, int out_size, void* d_ws, size_t ws_size,
                              hipStream_t stream)
{
  (void)in_sizes; (void)n_in; (void)out_size; (void)ws_size;
  const float* cost_memory = (const float*)d_in[0];
  const float* context     = (const float*)d_in[1];
  const float* cost_maps   = (const float*)d_in[2];
  // d_in[3] = query_latent_dim (scalar, unused)
  const float* proj_w   = (const float*)d_in[4];
  const float* proj_b   = (const float*)d_in[5];
  const float* fte_w1   = (const float*)d_in[6];
  const float* fte_b1   = (const float*)d_in[7];
  const float* fte_w2   = (const float*)d_in[8];
  const float* fte_b2   = (const float*)d_in[9];
  const float* ca_ln1_g = (const float*)d_in[10];
  const float* ca_ln1_b = (const float*)d_in[11];
  const float* ca_ln2_g = (const float*)d_in[12];
  const float* ca_ln2_b = (const float*)d_in[13];
  const float* ca_q_w   = (const float*)d_in[14];
  const float* ca_q_b   = (const float*)d_in[15];
  const float* ca_k_w   = (const float*)d_in[16];
  const float* ca_k_b   = (const float*)d_in[17];
  const float* ca_v_w   = (const float*)d_in[18];
  const float* ca_v_b   = (const float*)d_in[19];
  const float* ca_proj_w= (const float*)d_in[20];
  const float* ca_proj_b= (const float*)d_in[21];
  const float* ca_ffn_w1= (const float*)d_in[22];
  const float* ca_ffn_b1= (const float*)d_in[23];
  const float* ca_ffn_w2= (const float*)d_in[24];
  const float* ca_ffn_b2= (const float*)d_in[25];
  const float* att_qk_w = (const float*)d_in[26];
  const float* me_c1_w  = (const float*)d_in[27];
  const float* me_c1_b  = (const float*)d_in[28];
  const float* me_c2_w  = (const float*)d_in[29];
  const float* me_c2_b  = (const float*)d_in[30];
  const float* me_f1_w  = (const float*)d_in[31];
  const float* me_f1_b  = (const float*)d_in[32];
  const float* me_f2_w  = (const float*)d_in[33];
  const float* me_f2_b  = (const float*)d_in[34];
  const float* me_cv_w  = (const float*)d_in[35];
  const float* me_cv_b  = (const float*)d_in[36];
  const float* agg_v_w  = (const float*)d_in[37];
  const float* agg_gamma= (const float*)d_in[38];
  const float* gru_z1_w = (const float*)d_in[39];
  const float* gru_z1_b = (const float*)d_in[40];
  const float* gru_r1_w = (const float*)d_in[41];
  const float* gru_r1_b = (const float*)d_in[42];
  const float* gru_q1_w = (const float*)d_in[43];
  const float* gru_q1_b = (const float*)d_in[44];
  const float* gru_z2_w = (const float*)d_in[45];
  const float* gru_z2_b = (const float*)d_in[46];
  const float* gru_r2_w = (const float*)d_in[47];
  const float* gru_r2_b = (const float*)d_in[48];
  const float* gru_q2_w = (const float*)d_in[49];
  const float* gru_q2_b = (const float*)d_in[50];
  const float* fh1_w    = (const float*)d_in[51];
  const float* fh1_b    = (const float*)d_in[52];
  const float* fh2_w    = (const float*)d_in[53];
  const float* fh2_b    = (const float*)d_in[54];
  const float* mk1_w    = (const float*)d_in[55];
  const float* mk1_b    = (const float*)d_in[56];
  const float* mk2_w    = (const float*)d_in[57];
  const float* mk2_b    = (const float*)d_in[58];

  float* ws = (float*)d_ws;
  size_t off = 0;
  auto alloc = [&](size_t n) { float* p = ws + off; off += n; return p; };

  float* attnbuf = alloc((size_t)HW * HW);     // 3072x3072 self-attention
  float* keybuf  = alloc((size_t)NK * 64);
  float* valbuf  = alloc((size_t)NK * 64);
  float* net     = alloc(128 * HW);
  float* qbuf    = alloc(128 * HW);
  float* kbuf    = alloc(128 * HW);
  float* corr    = alloc(145 * HW);            // [cost_global(64) | cf(81)]
  float* fteh    = alloc(64 * HW);
  float* query   = alloc(64 * HW);
  float* qn_pe   = alloc(64 * HW);
  float* qproj   = alloc(64 * HW);
  float* cat128  = alloc(128 * HW);
  float* projout = alloc(64 * HW);
  float* xbuf    = alloc(64 * HW);
  float* xnbuf   = alloc(64 * HW);
  float* ffnh    = alloc(64 * HW);
  float* ffno    = alloc(64 * HW);
  float* cor256  = alloc(256 * HW);
  float* cat256  = alloc(256 * HW);            // [me_c2(192) | me_f2(64)]
  float* flo128  = alloc(128 * HW);
  float* flowbuf = alloc(2 * HW);
  float* xcat    = alloc(384 * HW);            // [inp(128)|mf(128)|mf_g(128)]
  float* vbuf    = alloc(128 * HW);
  float* glob    = alloc(128 * HW);
  float* zbuf    = alloc(128 * HW);
  float* rbuf    = alloc(128 * HW);
  float* rhbuf   = alloc(128 * HW);
  float* qg      = alloc(128 * HW);
  float* tmp256  = alloc(256 * HW);
  float* dfbuf   = alloc(2 * HW);
  float* maskbuf = alloc(576 * HW);
  float* coords1 = alloc(2 * HW);
  const float* inp = xcat;                     // inp written into xcat[0:128]

  // ---- setup phase ------------------------------------------------------
  L1D(k_init_coords, HW, coords1);
  // ctx projection, split: net = tanh(ctx[:128]), inp = relu(ctx[128:])
  gemm(stream, context, 1, HW, proj_w, 1, 256, net, 1, HW,
       proj_b, HW, 128, 256, 1.f, 3);
  gemm(stream, context, 1, HW, proj_w + 128 * 256, 1, 256, xcat, 1, HW,
       proj_b + 128, HW, 128, 256, 1.f, 1);
  // q/k 1x1 convs on inp (128 in-channels, att_qk_w is 256x128; no bias)
  gemm(stream, inp, 1, HW, att_qk_w, 1, 128, qbuf, 128, 1,
       nullptr, HW, 128, 128, 1.f, 0);
  gemm(stream, inp, 1, HW, att_qk_w + 128 * 128, 1, 128, kbuf, 128, 1,
       nullptr, HW, 128, 128, 1.f, 0);
  // attn = softmax(qf @ kf^T / sqrt(128))
  gemm(stream, qbuf, 128, 1, kbuf, 1, 128, attnbuf, HW, 1,
       nullptr, HW, HW, 128, 0.08838834764831845f, 0);
  softmax_rows<<<HW, 256, 0, stream>>>(attnbuf, HW);
  // key/value projections of cost_memory
  gemm(stream, cost_memory, 128, 1, ca_k_w, 64, 1, keybuf, 64, 1,
       ca_k_b, NK, 64, 128, 1.f, 0);
  gemm(stream, cost_memory, 128, 1, ca_v_w, 64, 1, valbuf, 64, 1,
       ca_v_b, NK, 64, 128, 1.f, 0);

  // ---- iterative refinement --------------------------------------------
  for (int it = 0; it < DEPTH; ++it) {
    // cost-volume sampling -> corr channels [64..145)
    L1D(cf_sample, 81 * HW, cost_maps, coords1, corr);
    // flow-token encoder (1x1 convs)
    gemm(stream, corr + (long)64 * HW, 1, HW, fte_w1, 1, 81, fteh, 64, 1,
         fte_b1, HW, 64, 81, 1.f, 2);
    gemm(stream, fteh, 64, 1, fte_w2, 1, 64, query, 64, 1,
         fte_b2, HW, 64, 64, 1.f, 0);
    // cross attention
    k_ln_pe<<<HW, 64, 0, stream>>>(query, ca_ln1_g, ca_ln1_b, coords1, qn_pe);
    gemm(stream, qn_pe, 64, 1, ca_q_w, 64, 1, qproj, 64, 1,
         ca_q_b, HW, 64, 64, 1.f, 0);
    attn_core<<<HW, 384, 0, stream>>>(qproj, keybuf, valbuf, query, cat128);
    gemm(stream, cat128, 128, 1, ca_proj_w, 64, 1, projout, 64, 1,
         ca_proj_b, HW, 64, 128, 1.f, 0);
    k_x_xn<<<HW, 64, 0, stream>>>(query, projout, ca_ln2_g, ca_ln2_b, xbuf, xnbuf);
    gemm(stream, xnbuf, 64, 1, ca_ffn_w1, 64, 1, ffnh, 64, 1,
         ca_ffn_b1, HW, 64, 64, 1.f, 2);
    gemm(stream, ffnh, 64, 1, ca_ffn_w2, 64, 1, ffno, 64, 1,
         ca_ffn_b2, HW, 64, 64, 1.f, 0);
    L1D(k_cost_global, 64 * HW, xbuf, ffno, corr);  // corr[0:64] = x + ffn

    // motion encoder
    gemm(stream, corr, 1, HW, me_c1_w, 1, 145, cor256, 1, HW,
         me_c1_b, HW, 256, 145, 1.f, 1);
    conv(stream, cor256, 256, nullptr, 0, me_c2_w, me_c2_b, cat256, 192, 3, 3, 1, 1, 1);
    L1D(k_flow, HW, coords1, flowbuf, xcat);
    conv(stream, flowbuf, 2, nullptr, 0, me_f1_w, me_f1_b, flo128, 128, 7, 7, 3, 3, 1);
    conv(stream, flo128, 128, nullptr, 0, me_f2_w, me_f2_b, cat256 + (long)192 * HW,
         64, 3, 3, 1, 1, 1);
    conv(stream, cat256, 256, nullptr, 0, me_cv_w, me_cv_b, xcat + (long)128 * HW,
         126, 3, 3, 1, 1, 1);                   // mf[0:126]; mf[126:128]=flow
    // global aggregation:  glob = attn @ (mf @ Wv)
    gemm(stream, xcat + (long)128 * HW, 1, HW, agg_v_w, 1, 128, vbuf, 128, 1,
         nullptr, HW, 128, 128, 1.f, 0);
    gemm(stream, attnbuf, HW, 1, vbuf, 128, 1, glob, 1, HW,
         nullptr, HW, 128, HW, 1.f, 0);
    k_mfg<<<(128 * HW + 255) / 256, 256, 0, stream>>>(xcat, glob, agg_gamma);

    // separable GRU, horizontal (1x5) then vertical (5x1)
    conv(stream, net, 128, xcat, 384, gru_z1_w, gru_z1_b, zbuf, 128, 1, 5, 0, 2, 4);
    conv(stream, net, 128, xcat, 384, gru_r1_w, gru_r1_b, rbuf, 128, 1, 5, 0, 2, 4);
    L1D(k_mul, 128 * HW, rbuf, net, rhbuf, 128 * HW);
    conv(stream, rhbuf, 128, xcat, 384, gru_q1_w, gru_q1_b, qg, 128, 1, 5, 0, 2, 3);
    L1D(k_gru, 128 * HW, net, zbuf, qg, 128 * HW);
    conv(stream, net, 128, xcat, 384, gru_z2_w, gru_z2_b, zbuf, 128, 5, 1, 2, 0, 4);
    conv(stream, net, 128, xcat, 384, gru_r2_w, gru_r2_b, rbuf, 128, 5, 1, 2, 0, 4);
    L1D(k_mul, 128 * HW, rbuf, net, rhbuf, 128 * HW);
    conv(stream, rhbuf, 128, xcat, 384, gru_q2_w, gru_q2_b, qg, 128, 5, 1, 2, 0, 3);
    L1D(k_gru, 128 * HW, net, zbuf, qg, 128 * HW);

    // flow head + mask head
    conv(stream, net, 128, nullptr, 0, fh1_w, fh1_b, tmp256, 256, 3, 3, 1, 1, 1);
    conv(stream, tmp256, 256, nullptr, 0, fh2_w, fh2_b, dfbuf, 2, 3, 3, 1, 1, 0);
    conv(stream, net, 128, nullptr, 0, mk1_w, mk1_b, tmp256, 256, 3, 3, 1, 1, 1);
    gemm(stream, tmp256, 1, HW, mk2_w, 1, 256, maskbuf, 1, HW,
         mk2_b, HW, 576, 256, 0.25f, 0);

    L1D(k_addcoords, 2 * HW, coords1, dfbuf);
    L1D(upsample_k, HW * 64, maskbuf, coords1,
        (float*)d_out + (size_t)it * 2 * 384 * 512);
  }
}